// InstanSeg_76802605187430
// MI455X (gfx1250) — compile-verified
//
#include <hip/hip_runtime.h>

// ---------------------------------------------------------------------------
// InstanSeg pipeline for MI455X (gfx1250, wave32)
//   K=256 instances, 128x128 windows, 34-ch features, MLP 34->128->128->1 (fp32)
//   - Layer GEMMs on V_WMMA_F32_16X16X4_F32 (CDNA5 fp32 matrix op, keeps ref precision)
//   - Operand staging into LDS via the Tensor Data Mover (tensor_load_to_lds)
//   - Affinity merge as a sparse gather (avg ~28 overlapping windows/pixel,
//     dist buffer is L2-resident) instead of a 300 MB dense scatter + GEMM.
// ---------------------------------------------------------------------------

typedef __attribute__((ext_vector_type(2))) float        v2f;
typedef __attribute__((ext_vector_type(8))) float        v8f;
typedef __attribute__((ext_vector_type(4))) unsigned int v4u;
typedef __attribute__((ext_vector_type(4))) int          v4i;
typedef __attribute__((ext_vector_type(8))) int          v8i;

#define E_CH    32
#define DIN     34
#define DIN_PAD 36
#define HID     128
#define KINST   256
#define HDIM    384
#define WDIM    384
#define HWSZ    (HDIM * WDIM)
#define WSZ     128
#define HALFW   64
#define NPIX    (WSZ * WSZ)        // 16384 pixels per window

__device__ __forceinline__ int clampc(int v) {
    return v < HALFW ? HALFW : (v > (HDIM - HALFW) ? (HDIM - HALFW) : v);
}

#if __has_builtin(__builtin_amdgcn_tensor_load_to_lds)
#define USE_TDM 1
#else
#define USE_TDM 0
#endif

#if USE_TDM
// ---------------------------------------------------------------------------
// Issue one TDM descriptor: copy a (tile_d1 x tile_d0) f32 tile (row stride
// stride0 elements) from global memory into contiguous LDS at lds_off.
// tile_d1 == 1 -> 1D copy. D# layout per CDNA5 ISA 8.3/8.4.
// ---------------------------------------------------------------------------
__device__ __forceinline__ void tdm_load_f32(unsigned lds_off, const float* gptr,
                                             unsigned tile_d0, unsigned tile_d1,
                                             unsigned long stride0) {
    const unsigned long ga = (unsigned long)gptr;
    v4u g0;
    g0.x = 1u;                                                 // count=1, user D#
    g0.y = lds_off;                                            // lds_addr (bytes)
    g0.z = (unsigned)(ga & 0xFFFFFFFFu);                       // global_addr[31:0]
    g0.w = (unsigned)((ga >> 32) & 0x1FFFFFFu) | (2u << 30);   // global_addr[56:32] | type=2
    v8i g1;
    g1[0] = (int)(2u << 16);                                   // data_size=4B; no multicast/pad/iter
    g1[1] = (int)((tile_d0 & 0xFFFFu) << 16);                  // tensor_dim0[15:0]
    g1[2] = (int)(((tile_d0 >> 16) & 0xFFFFu) |
                  ((tile_d1 & 0xFFFFu) << 16));                // tensor_dim0[31:16] | tensor_dim1[15:0]
    g1[3] = (int)(((tile_d1 >> 16) & 0xFFFFu) |
                  ((tile_d0 & 0xFFFFu) << 16));                // tensor_dim1[31:16] | tile_dim0
    g1[4] = (int)((tile_d1 > 1u ? tile_d1 : 0u) & 0xFFFFu);    // tile_dim1 (0 => 1D), tile_dim2=0
    g1[5] = (int)(unsigned)(stride0 & 0xFFFFFFFFul);           // tensor_dim0_stride[31:0]
    g1[6] = (int)(unsigned)((stride0 >> 32) & 0xFFFFul);       // stride[47:32] | dim1_stride[15:0]=0
    g1[7] = 0;
    const v4i gz = {0, 0, 0, 0};
#if defined(__clang_major__) && (__clang_major__ >= 23)
    const v8i gz8 = {0, 0, 0, 0, 0, 0, 0, 0};
    __builtin_amdgcn_tensor_load_to_lds(g0, g1, gz, gz, gz8, 0);
#else
    __builtin_amdgcn_tensor_load_to_lds(g0, g1, gz, gz, 0);
#endif
}
#endif

// ===========================================================================
// Kernel A: fused crop + per-pixel MLP -> dist[k, yy, xx]
//   grid = (256 instances, 128 rows), block = 256 threads = 8 wave32
//   Wave 0 drives the TDM; each wave owns 16 pixels (M) x 128 hidden (N).
// ===========================================================================
__launch_bounds__(256)
__global__ void mlp_dist_kernel(const float* __restrict__ x,
                                const float* __restrict__ sg,
                                const float* __restrict__ cvec,
                                const int*   __restrict__ cent,
                                const float* __restrict__ W1,
                                const float* __restrict__ b1,
                                const float* __restrict__ W2,
                                const float* __restrict__ b2,
                                const float* __restrict__ W3,
                                const float* __restrict__ b3,
                                float* __restrict__ dist)
{
    // LDS staging (~151 KB of the 320 KB/WGP):
    __shared__ float sBuf[8 * 16 * HID];     // 64 KB: feats[36x128], then h1[128px x 128]
    __shared__ float sW1[DIN_PAD * HID];     // 18.4 KB (K padded 34->36)
    __shared__ float sW2[HID * HID];         // 64 KB
    __shared__ float sB1[HID];
    __shared__ float sB2[HID];
    __shared__ float sW3[HID];
    __shared__ float sC[DIN_PAD];            // c[k,:] (zero for sigma/pad channels)

    const int k  = blockIdx.x;
    const int yy = blockIdx.y;
    const int t  = threadIdx.x;

    const int cy = cent[2 * k], cx = cent[2 * k + 1];
    const int ty = clampc(cy) - HALFW;
    const int tx = clampc(cx) - HALFW;
    const int py = ty + yy;
    const long rowoff = (long)py * WDIM + tx;

#if USE_TDM
    // -------- async staging through the Tensor Data Mover (wave 0 only) ----
    if (t < 32) {
        // x crop rows: 2D tile 32 rows x 128 cols, row stride = H*W
        tdm_load_f32((unsigned)(unsigned long)(void*)&sBuf[0],
                     x + rowoff, WSZ, E_CH, (unsigned long)HWSZ);
        // sigma crop rows: 2 x 128
        tdm_load_f32((unsigned)(unsigned long)(void*)&sBuf[E_CH * WSZ],
                     sg + rowoff, WSZ, 2, (unsigned long)HWSZ);
        // W1: contiguous 34*128
        tdm_load_f32((unsigned)(unsigned long)(void*)&sW1[0],
                     W1, DIN * HID, 1, (unsigned long)(DIN * HID));
        // W2: contiguous 128*128
        tdm_load_f32((unsigned)(unsigned long)(void*)&sW2[0],
                     W2, HID * HID, 1, (unsigned long)(HID * HID));
#if __has_builtin(__builtin_amdgcn_s_wait_tensorcnt)
        __builtin_amdgcn_s_wait_tensorcnt(0);
#else
        asm volatile("s_wait_tensorcnt 0x0" ::: "memory");
#endif
    }
    // small fills in parallel with the TDM (disjoint LDS ranges)
    if (t < HID) { sB1[t] = b1[t]; sB2[t] = b2[t]; sW3[t] = W3[t]; }
    if (t >= 128 && t < 128 + DIN_PAD) {
        const int cc = t - 128;
        sC[cc] = (cc < E_CH) ? cvec[k * E_CH + cc] : 0.f;
    }
    // zero pad rows (feats rows 34..35, W1 rows 34..35)
    for (int i = t; i < 2 * WSZ; i += 256) sBuf[DIN * WSZ + i] = 0.f;
    for (int i = t; i < 2 * HID; i += 256) sW1[DIN * HID + i]  = 0.f;
#else
    // -------- fallback: VALU staging --------
    for (int i = t; i < DIN_PAD * WSZ; i += 256) {
        const int cch = i >> 7, px = i & 127;
        float v = 0.f;
        if (cch < E_CH)     v = x[(long)cch * HWSZ + rowoff + px];
        else if (cch < DIN) v = sg[(long)(cch - E_CH) * HWSZ + rowoff + px];
        sBuf[i] = v;
    }
    for (int i = t; i < DIN_PAD * HID; i += 256) sW1[i] = ((i >> 7) < DIN) ? W1[i] : 0.f;
    for (int i = t; i < HID * HID; i += 256) sW2[i] = W2[i];
    if (t < HID) { sB1[t] = b1[t]; sB2[t] = b2[t]; sW3[t] = W3[t]; }
    if (t >= 128 && t < 128 + DIN_PAD) {
        const int cc = t - 128;
        sC[cc] = (cc < E_CH) ? cvec[k * E_CH + cc] : 0.f;
    }
#endif
    __syncthreads();

    const int wv   = t >> 5;          // wave id 0..7
    const int lane = t & 31;
    const int half = lane >> 4;       // 0: lanes 0-15, 1: lanes 16-31
    const int lm   = lane & 15;
    const int px0  = wv * 16;         // this wave's 16 pixels

    const v8f VZ = {0.f, 0.f, 0.f, 0.f, 0.f, 0.f, 0.f, 0.f};
    v8f acc[8];

    // =============== layer 1: (16 x 36) @ (36 x 128) ===============
#pragma unroll
    for (int nt = 0; nt < 8; ++nt) acc[nt] = VZ;
#pragma unroll
    for (int ks = 0; ks < DIN_PAD / 4; ++ks) {
        const int kk = 4 * ks + 2 * half;   // A: lanes 0-15 K=k0,k0+1; lanes 16-31 K=k0+2,k0+3
        v2f a;
        a.x = sBuf[(kk    ) * WSZ + px0 + lm] - sC[kk];
        a.y = sBuf[(kk + 1) * WSZ + px0 + lm] - sC[kk + 1];
#pragma unroll
        for (int nt = 0; nt < 8; ++nt) {
            v2f b;
            b.x = sW1[(kk    ) * HID + nt * 16 + lm];
            b.y = sW1[(kk + 1) * HID + nt * 16 + lm];
            acc[nt] = __builtin_amdgcn_wmma_f32_16x16x4_f32(
                false, a, false, b, (short)0, acc[nt], false, false);
        }
    }

    __syncthreads();   // all waves done reading feats before sBuf is reused for h1

    // relu + bias; transpose C-layout -> row-major h1[pixel][n] in sBuf
#pragma unroll
    for (int nt = 0; nt < 8; ++nt) {
        const int n = nt * 16 + lm;
        const float bb = sB1[n];
#pragma unroll
        for (int r = 0; r < 8; ++r) {
            const int m = r + 8 * half;     // C layout: lanes0-15 M=r, lanes16-31 M=8+r
            const float h = acc[nt][r] + bb;
            sBuf[(px0 + m) * HID + n] = h > 0.f ? h : 0.f;
        }
    }
    // h1 is produced and consumed by the same wave: per-wave LDS ordering suffices.

    // =============== layer 2: (16 x 128) @ (128 x 128) ===============
#pragma unroll
    for (int nt = 0; nt < 8; ++nt) acc[nt] = VZ;
#pragma unroll 4
    for (int ks = 0; ks < HID / 4; ++ks) {
        const int kk = 4 * ks + 2 * half;
        v2f a;
        a.x = sBuf[(px0 + lm) * HID + kk];
        a.y = sBuf[(px0 + lm) * HID + kk + 1];
#pragma unroll
        for (int nt = 0; nt < 8; ++nt) {
            v2f b;
            b.x = sW2[(kk    ) * HID + nt * 16 + lm];
            b.y = sW2[(kk + 1) * HID + nt * 16 + lm];
            acc[nt] = __builtin_amdgcn_wmma_f32_16x16x4_f32(
                false, a, false, b, (short)0, acc[nt], false, false);
        }
    }

    // =============== layer 3: dist = relu(h2) . W3 + b3 ===============
    float part[8];
#pragma unroll
    for (int r = 0; r < 8; ++r) part[r] = 0.f;
#pragma unroll
    for (int nt = 0; nt < 8; ++nt) {
        const int n = nt * 16 + lm;
        const float w3v = sW3[n];
        const float bb  = sB2[n];
#pragma unroll
        for (int r = 0; r < 8; ++r) {
            const float h = acc[nt][r] + bb;
            part[r] += (h > 0.f ? h : 0.f) * w3v;
        }
    }
    const float b3v = b3[0];
#pragma unroll
    for (int r = 0; r < 8; ++r) {
        // reduce across the 16 lanes of each half (masks < 16 stay in-group)
        part[r] += __shfl_xor(part[r], 8, 32);
        part[r] += __shfl_xor(part[r], 4, 32);
        part[r] += __shfl_xor(part[r], 2, 32);
        part[r] += __shfl_xor(part[r], 1, 32);
        if (lm == 0) {
            const int m = r + 8 * half;
            dist[(long)k * NPIX + yy * WSZ + px0 + m] = part[r] + b3v;
        }
    }
}

// ===========================================================================
// Kernel B0: grab the centroid-pixel value per instance
// ===========================================================================
__global__ void center_kernel(const int* __restrict__ cent,
                              const float* __restrict__ dist,
                              float* __restrict__ centers)
{
    const int k = threadIdx.x;
    const int cy = cent[2 * k], cx = cent[2 * k + 1];
    const int ty = clampc(cy) - HALFW, tx = clampc(cx) - HALFW;
    const int cyi = cy - ty, cxi = cx - tx;
    centers[k] = dist[(long)k * NPIX + cyi * WSZ + cxi];
}

// ===========================================================================
// Kernel B1: in-place stable  d = a + b - log(1 + e^a + e^b)
//   (softplus(logaddexp(a,b)) == log(1 + e^a + e^b))
// ===========================================================================
__global__ void transform_kernel(float* __restrict__ dist,
                                 const float* __restrict__ centers)
{
    const int k   = blockIdx.x;
    const int off = blockIdx.y * 256 + threadIdx.x;
    const long idx = (long)k * NPIX + off;
    const float a = dist[idx];
    const float b = centers[k];
    const float m = fmaxf(fmaxf(a, b), 0.f);
    const float lse = m + __logf(__expf(a - m) + __expf(b - m) + __expf(-m));
    dist[idx] = a + b - lse;
}

// ===========================================================================
// Kernel C: affinity merge as a sparse gather.
//   merged(k,p) = sum_j Ms[k,j]*dist_j(p)*[p in win_j] / max(sum_j Ms[k,j]*[p in win_j], eps)
//   grid = (256 instances, 128 rows), block = 128 threads (one output row).
// ===========================================================================
__launch_bounds__(128)
__global__ void merge_kernel(const int* __restrict__ cent,
                             const float* __restrict__ Mmat,
                             const float* __restrict__ dist,
                             float* __restrict__ out)
{
    __shared__ float sMs[KINST];
    __shared__ int   sTy[KINST];
    __shared__ int   sTx[KINST];

    const int k  = blockIdx.x;
    const int yy = blockIdx.y;
    const int t  = threadIdx.x;

    for (int j = t; j < KINST; j += 128) {
        const int jcy = clampc(cent[2 * j]), jcx = clampc(cent[2 * j + 1]);
        sTy[j] = jcy - HALFW;
        sTx[j] = jcx - HALFW;
        sMs[j] = (j == k) ? 1.f : 0.5f * (Mmat[k * KINST + j] + Mmat[j * KINST + k]);
    }
    __syncthreads();

    const int py  = sTy[k] + yy;
    const int pxg = sTx[k] + t;
    float num = 0.f, den = 0.f;
#pragma unroll 4
    for (int j = 0; j < KINST; ++j) {
        const unsigned dy = (unsigned)(py - sTy[j]);
        const unsigned dx = (unsigned)(pxg - sTx[j]);
        if (dy < WSZ && dx < WSZ) {
            const float ms = sMs[j];
            num += ms * dist[(long)j * NPIX + dy * WSZ + dx];
            den += ms;
        }
    }
    out[(long)k * NPIX + yy * WSZ + t] = num / fmaxf(den, 1e-6f);
}

// ===========================================================================
// Host-side launcher. Inputs in setup_inputs() order:
//   0:x 1:sigma 2:c 3:centroids(int) 4:M 5:W1 6:b1 7:W2 8:b2 9:W3 10:b3
// Workspace: [0, K*NPIX) dist fp32, then [K*NPIX, K*NPIX+256) centers.
// ===========================================================================
extern "C" void kernel_launch(void* const* d_in, const int* in_sizes, int n_in,
                              void* d_out, int out_size, void* d_ws, size_t ws_size,
                              hipStream_t stream) {
    const float* x    = (const float*)d_in[0];
    const float* sg   = (const float*)d_in[1];
    const float* cvec = (const float*)d_in[2];
    const int*   cent = (const int*)  d_in[3];
    const float* Mm   = (const float*)d_in[4];
    const float* W1   = (const float*)d_in[5];
    const float* b1   = (const float*)d_in[6];
    const float* W2   = (const float*)d_in[7];
    const float* b2   = (const float*)d_in[8];
    const float* W3   = (const float*)d_in[9];
    const float* b3   = (const float*)d_in[10];
    float* out = (float*)d_out;

    float* dist    = (float*)d_ws;                 // K * 16384 floats
    float* centers = dist + (long)KINST * NPIX;    // 256 floats

    mlp_dist_kernel<<<dim3(KINST, WSZ), 256, 0, stream>>>(
        x, sg, cvec, cent, W1, b1, W2, b2, W3, b3, dist);
    center_kernel<<<1, KINST, 0, stream>>>(cent, dist, centers);
    transform_kernel<<<dim3(KINST, NPIX / 256), 256, 0, stream>>>(dist, centers);
    merge_kernel<<<dim3(KINST, WSZ), 128, 0, stream>>>(cent, Mm, dist, out);
}